// SAGECluster_29137058136186
// MI455X (gfx1250) — compile-verified
//
#include <hip/hip_runtime.h>
#include <hip/hip_bf16.h>
#include <cstdint>

typedef __attribute__((ext_vector_type(2))) float v2f;
typedef __attribute__((ext_vector_type(8))) float v8f;

#define NODES 50000
#define D0 128
#define D1 256
#define D2 128

// ---------------------------------------------------------------------------
// init: agg = src feature copy (self-loop contribution); optionally deg = 1
// one float4 per thread over N*D/4 elements
// ---------------------------------------------------------------------------
__global__ __launch_bounds__(256) void init_copy_kernel(
    const float* __restrict__ x, float* __restrict__ agg,
    float* __restrict__ deg, int n4, int n) {
  int t = blockIdx.x * blockDim.x + threadIdx.x;
  if (t < n4) {
    ((float4*)agg)[t] = ((const float4*)x)[t];
  }
  if (deg != nullptr && t < n) deg[t] = 1.0f;
}

// ---------------------------------------------------------------------------
// degree scatter: one thread per edge
// ---------------------------------------------------------------------------
__global__ __launch_bounds__(256) void deg_scatter_kernel(
    const long long* __restrict__ ei, int E, float* __restrict__ deg) {
  int e = blockIdx.x * blockDim.x + threadIdx.x;
  if (e >= E) return;
  int d = (int)ei[(long long)E + e];
  atomicAdd(&deg[d], 1.0f);
}

__global__ __launch_bounds__(256) void recip_kernel(float* __restrict__ deg, int n) {
  int t = blockIdx.x * blockDim.x + threadIdx.x;
  if (t < n) deg[t] = 1.0f / deg[t];
}

// ---------------------------------------------------------------------------
// feature scatter-add over edges: one wave (32 lanes) per edge, float4/lane
// agg[dst] += feat[src]  (128 floats per edge)
// ---------------------------------------------------------------------------
__global__ __launch_bounds__(256) void edge_scatter128_kernel(
    const long long* __restrict__ ei, int E,
    const float* __restrict__ feat, float* __restrict__ agg) {
  long long g = (long long)blockIdx.x * blockDim.x + threadIdx.x;
  int e    = (int)(g >> 5);
  int lane = (int)(g & 31);
  if (e >= E) return;
  int s = (int)ei[e];
  int d = (int)ei[(long long)E + e];
  const float4 v = *(const float4*)(feat + (size_t)s * 128 + lane * 4);
  float* p = agg + (size_t)d * 128 + lane * 4;
  atomicAdd(p + 0, v.x);
  atomicAdd(p + 1, v.y);
  atomicAdd(p + 2, v.z);
  atomicAdd(p + 3, v.w);
}

// ---------------------------------------------------------------------------
// WMMA GEMM: C[M x Nout] = act( (A * rowScale?) @ W^T + bias? )
//   W is [Nout x K] row-major (PyTorch linear layout), so B[k][n] = W[n][k].
// Block: 256 threads = 8 waves; each wave owns one 16x16 output tile;
// block covers 16 rows x 128 cols. A tile (16 x K) staged in LDS.
// Uses V_WMMA_F32_16X16X4_F32 (full f32 precision on the matrix pipe).
// Per ISA layouts (wave32):
//   A frag: lane half h = lane>>4, m = lane&15 : a[v] = A[m][k0 + 2h + v]
//   B frag (mirror, lanes index N):             b[v] = B[k0+2h+v][n] = W[n][k0+2h+v]
//   C/D  : c[v] -> row (v + 8h), col = lane&15
// ---------------------------------------------------------------------------
template <int K, bool SCALE_A, bool HAS_BIAS, bool RELU>
__global__ __launch_bounds__(256) void wmma_gemm_kernel(
    const float* __restrict__ A, const float* __restrict__ W,
    const float* __restrict__ bias, const float* __restrict__ rowScale,
    float* __restrict__ C, int Nout) {
  constexpr int LDA = K + 4;                 // pad: row stride 528B/1040B, 16B aligned,
  __shared__ float As[16 * LDA];             // and banks spread across the 16 lanes

  const int tid  = threadIdx.x;
  const int wave = tid >> 5;
  const int lane = tid & 31;
  const int half = lane >> 4;
  const int m16  = lane & 15;
  const int mBase = blockIdx.x * 16;
  const int nBase = blockIdx.y * 128 + wave * 16;

  // cooperative load of the 16 x K A-tile (scaled by inv_deg if requested)
  constexpr int NV4 = 16 * K / 4;
  for (int idx = tid; idx < NV4; idx += 256) {
    int r  = idx / (K / 4);
    int cv = idx % (K / 4);
    float4 v = *(const float4*)(A + (size_t)(mBase + r) * K + cv * 4);
    if (SCALE_A) {
      float s = rowScale[mBase + r];
      v.x *= s; v.y *= s; v.z *= s; v.w *= s;
    }
    *(float4*)(&As[r * LDA + cv * 4]) = v;
  }
  __syncthreads();

  v8f acc = {};
  const float* Wn   = W + (size_t)(nBase + m16) * K;   // this lane's output column of W
  const float* Arow = &As[m16 * LDA];
  const int koff = half * 2;

#pragma unroll
  for (int k0 = 0; k0 < K; k0 += 4) {
    v2f a, b;
    a.x = Arow[k0 + koff];
    a.y = Arow[k0 + koff + 1];
    b.x = Wn[k0 + koff];
    b.y = Wn[k0 + koff + 1];
    acc = __builtin_amdgcn_wmma_f32_16x16x4_f32(
        /*neg_a=*/false, a, /*neg_b=*/false, b,
        /*c_mod=*/(short)0, acc, /*reuse_a=*/false, /*reuse_b=*/false);
  }

  const int col = nBase + m16;
  float bv = HAS_BIAS ? bias[col] : 0.0f;
#pragma unroll
  for (int v = 0; v < 8; ++v) {
    int row = mBase + v + half * 8;
    float val = acc[v] + bv;
    if (RELU) val = fmaxf(val, 0.0f);
    C[(size_t)row * Nout + col] = val;
  }
}

// ---------------------------------------------------------------------------
// finalize: v = agg2[i]*inv_deg[i] + b2 ; out = v / max(||v||,1e-12) * exp(ls)
// one wave per node, float4 per lane, shfl_xor wave reduction
// ---------------------------------------------------------------------------
__global__ __launch_bounds__(256) void finalize_kernel(
    const float* __restrict__ agg2, const float* __restrict__ invdeg,
    const float* __restrict__ b2, const float* __restrict__ logit_scale,
    float* __restrict__ out, int n) {
  long long g = (long long)blockIdx.x * blockDim.x + threadIdx.x;
  int i    = (int)(g >> 5);
  int lane = (int)(g & 31);
  if (i >= n) return;

  float s = invdeg[i];
  float4 v  = *(const float4*)(agg2 + (size_t)i * 128 + lane * 4);
  float4 bb = *(const float4*)(b2 + lane * 4);
  v.x = v.x * s + bb.x;
  v.y = v.y * s + bb.y;
  v.z = v.z * s + bb.z;
  v.w = v.w * s + bb.w;

  float ss = v.x * v.x + v.y * v.y + v.z * v.z + v.w * v.w;
#pragma unroll
  for (int off = 16; off > 0; off >>= 1) ss += __shfl_xor(ss, off, 32);

  float nrm = fmaxf(sqrtf(ss), 1e-12f);
  float sc  = expf(logit_scale[0]) / nrm;

  float4 o = make_float4(v.x * sc, v.y * sc, v.z * sc, v.w * sc);
  *(float4*)(out + (size_t)i * 128 + lane * 4) = o;
}

// ---------------------------------------------------------------------------
// launch
// inputs: 0:x(N*128) 1:edge_index(2*E int64) 2:W1(256*128) 3:b1(256)
//         4:W2(128*256) 5:b2(128) 6:logit_scale(1)
// ---------------------------------------------------------------------------
extern "C" void kernel_launch(void* const* d_in, const int* in_sizes, int n_in,
                              void* d_out, int out_size, void* d_ws, size_t ws_size,
                              hipStream_t stream) {
  const float*     x  = (const float*)d_in[0];
  const long long* ei = (const long long*)d_in[1];
  const float*     W1 = (const float*)d_in[2];
  const float*     b1 = (const float*)d_in[3];
  const float*     W2 = (const float*)d_in[4];
  const float*     b2 = (const float*)d_in[5];
  const float*     ls = (const float*)d_in[6];
  float* out = (float*)d_out;

  const int N = in_sizes[0] / D0;   // 50000
  const int E = in_sizes[1] / 2;    // 640000

  // workspace partition (all f32): deg | agg1(reused as agg2) | y1 | z
  float* deg  = (float*)d_ws;               // N
  float* agg1 = deg  + N;                   // N*128  (N*4 bytes keeps 16B align)
  float* y1   = agg1 + (size_t)N * D0;      // N*256
  float* z    = y1   + (size_t)N * D1;      // N*128

  const int n4 = N * (D0 / 4);                       // float4 count for 128-dim copy
  dim3 blk(256);
  dim3 gCopy((n4 + 255) / 256);
  dim3 gEdge1((E + 255) / 256);
  dim3 gEdge32(((long long)E * 32 + 255) / 256);
  dim3 gNode32(((long long)N * 32 + 255) / 256);

  // ---- layer 1 aggregation: agg1 = x (self loop) + sum_{e} x[src[e]] at dst
  init_copy_kernel<<<gCopy, blk, 0, stream>>>(x, agg1, deg, n4, N);
  deg_scatter_kernel<<<gEdge1, blk, 0, stream>>>(ei, E, deg);
  recip_kernel<<<(N + 255) / 256, blk, 0, stream>>>(deg, N);   // deg -> inv_deg
  edge_scatter128_kernel<<<gEdge32, blk, 0, stream>>>(ei, E, x, agg1);

  // ---- y1 = relu((agg1 * inv_deg) @ W1^T + b1)   [50000x128 @ 128x256]
  wmma_gemm_kernel<D0, true, true, true>
      <<<dim3(N / 16, D1 / 128), blk, 0, stream>>>(agg1, W1, b1, deg, y1, D1);

  // ---- z = y1 @ W2^T  (bias applied after aggregation)   [50000x256 @ 256x128]
  wmma_gemm_kernel<D1, false, false, false>
      <<<dim3(N / 16, D2 / 128), blk, 0, stream>>>(y1, W2, nullptr, nullptr, z, D2);

  // ---- layer 2 aggregation over z (128-dim, reuse agg1 buffer as agg2)
  init_copy_kernel<<<gCopy, blk, 0, stream>>>(z, agg1, nullptr, n4, N);
  edge_scatter128_kernel<<<gEdge32, blk, 0, stream>>>(ei, E, z, agg1);

  // ---- out = normalize(agg2*inv_deg + b2) * exp(ls)
  finalize_kernel<<<gNode32, blk, 0, stream>>>(agg1, deg, b2, ls, out, N);
}